// DeepGCN_15453292331027
// MI455X (gfx1250) — compile-verified
//
#include <hip/hip_runtime.h>
#include <hip/hip_bf16.h>

typedef __attribute__((ext_vector_type(2))) float v2f;
typedef __attribute__((ext_vector_type(8))) float v8f;

#define HDIM 128
#define HDIM2 256
#define NUM_GRAPHS 64
#define NUM_CLASSES 10
#define MSG_EPS 1e-7f

// ---------------------------------------------------------------------------
// exact float atomic max: sign-split into int max / uint min (order-invariant)
// ---------------------------------------------------------------------------
__device__ __forceinline__ void atomicMaxF(float* addr, float v) {
  if (v >= 0.0f) atomicMax((int*)addr, __float_as_int(v));
  else           atomicMin((unsigned int*)addr, __float_as_uint(v));
}

// ---------------------------------------------------------------------------
// GEMM: Y[N,M] = (accum ? Y : 0) + X[N,K] @ W[K,M] + bias[M]
// One wave per 16x16 tile, K stepped by 4 via v_wmma_f32_16x16x4_f32.
// A 16x4 layout: VGPR0 = {K0 lanes0-15, K2 lanes16-31}, VGPR1 = {K1, K3}
// B 4x16 layout mirrored; C/D = documented 8-VGPR striping.
// ---------------------------------------------------------------------------
__global__ void gemm16_wmma_f32(const float* __restrict__ X,
                                const float* __restrict__ W,
                                const float* __restrict__ bias,
                                float* __restrict__ Y,
                                int Kdim, int Mdim, int accum) {
  const int lane = threadIdx.x & 31;
  const int wid  = threadIdx.x >> 5;                 // 0..3 waves per block
  const int row0 = blockIdx.x * 16;
  const int col0 = (blockIdx.y * 4 + wid) * 16;
  if (col0 >= Mdim) return;                          // wave-uniform

  const int half = lane >> 4;                        // 0: lanes 0-15, 1: 16-31
  const int l16  = lane & 15;

  const float bv = bias[col0 + l16];
  v8f c;
  if (accum) {
#pragma unroll
    for (int r = 0; r < 8; ++r)
      c[r] = Y[(size_t)(row0 + half * 8 + r) * Mdim + col0 + l16] + bv;
  } else {
#pragma unroll
    for (int r = 0; r < 8; ++r) c[r] = bv;
  }

  const float* xrow = X + (size_t)(row0 + l16) * Kdim + half * 2;
  const float* wcol = W + (size_t)(half * 2) * Mdim + col0 + l16;

  for (int k = 0; k < Kdim; k += 4) {
    __builtin_prefetch(xrow + k + 32, 0, 1);         // global_prefetch_b8
    float2 av = *(const float2*)(xrow + k);          // K = k+2*half, +1
    v2f a; a[0] = av.x; a[1] = av.y;
    v2f b;
    b[0] = wcol[(size_t)k * Mdim];                   // K = k+2*half
    b[1] = wcol[(size_t)(k + 1) * Mdim];             // K = k+2*half+1
    c = __builtin_amdgcn_wmma_f32_16x16x4_f32(false, a, false, b,
                                              (short)0, c, false, false);
  }

#pragma unroll
  for (int r = 0; r < 8; ++r)
    Y[(size_t)(row0 + half * 8 + r) * Mdim + col0 + l16] = c[r];
}

// ---------------------------------------------------------------------------
// LayerNorm (+ReLU) over rows of width D; one wave per row (wave32 shuffles)
// ---------------------------------------------------------------------------
__global__ void ln_relu_kernel(const float* __restrict__ x, float* __restrict__ y,
                               const float* __restrict__ g, const float* __restrict__ b,
                               int rows, int D) {
  int row  = blockIdx.x * (blockDim.x >> 5) + (threadIdx.x >> 5);
  int lane = threadIdx.x & 31;
  if (row >= rows) return;
  const float* xr = x + (size_t)row * D;
  float s = 0.f, q = 0.f;
  for (int j = lane; j < D; j += 32) { float v = xr[j]; s += v; q += v * v; }
#pragma unroll
  for (int off = 16; off > 0; off >>= 1) {
    s += __shfl_xor(s, off, 32);
    q += __shfl_xor(q, off, 32);
  }
  float mean = s / (float)D;
  float var  = q / (float)D - mean * mean;
  float inv  = rsqrtf(var + 1e-5f);
  float* yr = y + (size_t)row * D;
  for (int j = lane; j < D; j += 32) {
    float v = (xr[j] - mean) * inv * g[j] + b[j];
    yr[j] = fmaxf(v, 0.f);
  }
}

// ---------------------------------------------------------------------------
// per-layer segment buffers init: mx = -inf, den = num = 0
// ---------------------------------------------------------------------------
__global__ void init_seg_kernel(float* mx, float* den, float* num, int total) {
  int i = blockIdx.x * blockDim.x + threadIdx.x;
  if (i >= total) return;
  mx[i]  = __int_as_float(0xff800000);   // -inf
  den[i] = 0.f;
  num[i] = 0.f;
}

__global__ void zero_kernel(float* p, int n) {
  int i = blockIdx.x * blockDim.x + threadIdx.x;
  if (i < n) p[i] = 0.f;
}

// ---------------------------------------------------------------------------
// Edge pass 1: m = relu(x[src]+e)+eps; atomic segment-max of m*t into mx[dst]
// One wave per edge; each lane owns a float4 chunk of the 128-wide row.
// ---------------------------------------------------------------------------
__global__ void edge_max_kernel(const float* __restrict__ x, const float* __restrict__ ea,
                                const int* __restrict__ src, const int* __restrict__ dst,
                                const float* __restrict__ tptr, int layer,
                                float* __restrict__ mx, int E) {
  int e    = blockIdx.x * (blockDim.x >> 5) + (threadIdx.x >> 5);
  int lane = threadIdx.x & 31;
  if (e >= E) return;
  int s = src[e], d = dst[e];
  float tv = tptr[layer];
  float4 a  = ((const float4*)(x  + (size_t)s * HDIM))[lane];
  float4 bb = ((const float4*)(ea + (size_t)e * HDIM))[lane];
  float m0 = fmaxf(a.x + bb.x, 0.f) + MSG_EPS;
  float m1 = fmaxf(a.y + bb.y, 0.f) + MSG_EPS;
  float m2 = fmaxf(a.z + bb.z, 0.f) + MSG_EPS;
  float m3 = fmaxf(a.w + bb.w, 0.f) + MSG_EPS;
  float* mrow = mx + (size_t)d * HDIM + lane * 4;
  atomicMaxF(mrow + 0, m0 * tv);
  atomicMaxF(mrow + 1, m1 * tv);
  atomicMaxF(mrow + 2, m2 * tv);
  atomicMaxF(mrow + 3, m3 * tv);
}

// ---------------------------------------------------------------------------
// Edge pass 2: ex = exp(m*t - mx[dst]); den[dst]+=ex; num[dst]+=m*ex
// (alpha normalization folds out: agg = num/den since den is segment-constant)
// ---------------------------------------------------------------------------
__global__ void edge_sum_kernel(const float* __restrict__ x, const float* __restrict__ ea,
                                const int* __restrict__ src, const int* __restrict__ dst,
                                const float* __restrict__ tptr, int layer,
                                const float* __restrict__ mx,
                                float* __restrict__ den, float* __restrict__ num, int E) {
  int e    = blockIdx.x * (blockDim.x >> 5) + (threadIdx.x >> 5);
  int lane = threadIdx.x & 31;
  if (e >= E) return;
  int s = src[e], d = dst[e];
  float tv = tptr[layer];
  float4 a  = ((const float4*)(x  + (size_t)s * HDIM))[lane];
  float4 bb = ((const float4*)(ea + (size_t)e * HDIM))[lane];
  float4 mv = ((const float4*)(mx + (size_t)d * HDIM))[lane];
  float m0 = fmaxf(a.x + bb.x, 0.f) + MSG_EPS;
  float m1 = fmaxf(a.y + bb.y, 0.f) + MSG_EPS;
  float m2 = fmaxf(a.z + bb.z, 0.f) + MSG_EPS;
  float m3 = fmaxf(a.w + bb.w, 0.f) + MSG_EPS;
  float e0 = __expf(m0 * tv - mv.x);
  float e1 = __expf(m1 * tv - mv.y);
  float e2 = __expf(m2 * tv - mv.z);
  float e3 = __expf(m3 * tv - mv.w);
  float* drow = den + (size_t)d * HDIM + lane * 4;
  float* nrow = num + (size_t)d * HDIM + lane * 4;
  atomicAdd(drow + 0, e0);       atomicAdd(nrow + 0, e0 * m0);
  atomicAdd(drow + 1, e1);       atomicAdd(nrow + 1, e1 * m1);
  atomicAdd(drow + 2, e2);       atomicAdd(nrow + 2, e2 * m2);
  atomicAdd(drow + 3, e3);       atomicAdd(nrow + 3, e3 * m3);
}

// ---------------------------------------------------------------------------
// hc = (den>0 ? num/den : 0) + r        (GENConv root skip)
// ---------------------------------------------------------------------------
__global__ void combine_kernel(const float* __restrict__ rin, const float* __restrict__ num,
                               const float* __restrict__ den, float* __restrict__ hc, int total) {
  int i = blockIdx.x * blockDim.x + threadIdx.x;
  if (i >= total) return;
  float dv = den[i];
  float agg = dv > 0.f ? num[i] / dv : 0.f;
  hc[i] = agg + rin[i];
}

// ---------------------------------------------------------------------------
// global_add_pool into pooled[64,128]
// ---------------------------------------------------------------------------
__global__ void pool_kernel(const float* __restrict__ x, const int* __restrict__ batch,
                            float* __restrict__ pooled, int n) {
  int i = blockIdx.x * blockDim.x + threadIdx.x;
  if (i >= n * HDIM) return;
  int node = i >> 7, col = i & (HDIM - 1);
  atomicAdd(pooled + (size_t)batch[node] * HDIM + col, x[i]);
}

// ---------------------------------------------------------------------------
// logits[64,10] = pooled @ lin_W + lin_b   (tiny: plain VALU)
// ---------------------------------------------------------------------------
__global__ void final_linear_kernel(const float* __restrict__ pooled,
                                    const float* __restrict__ W,
                                    const float* __restrict__ b,
                                    float* __restrict__ out) {
  int i = blockIdx.x * blockDim.x + threadIdx.x;
  if (i >= NUM_GRAPHS * NUM_CLASSES) return;
  int g = i / NUM_CLASSES, c = i % NUM_CLASSES;
  float acc = b[c];
  const float* pr = pooled + (size_t)g * HDIM;
#pragma unroll 4
  for (int k = 0; k < HDIM; ++k) acc += pr[k] * W[k * NUM_CLASSES + c];
  out[i] = acc;
}

// ---------------------------------------------------------------------------
extern "C" void kernel_launch(void* const* d_in, const int* in_sizes, int n_in,
                              void* d_out, int out_size, void* d_ws, size_t ws_size,
                              hipStream_t stream) {
  const float* x     = (const float*)d_in[0];
  const float* ea    = (const float*)d_in[1];
  const float* enc_W = (const float*)d_in[2];
  const float* enc_b = (const float*)d_in[3];
  const float* t     = (const float*)d_in[4];
  const float* W1    = (const float*)d_in[5];
  const float* b1    = (const float*)d_in[6];
  const float* mlp_g = (const float*)d_in[7];
  const float* mlp_b = (const float*)d_in[8];
  const float* W2    = (const float*)d_in[9];
  const float* b2    = (const float*)d_in[10];
  const float* ln_g  = (const float*)d_in[11];
  const float* ln_b  = (const float*)d_in[12];
  const float* lin_W = (const float*)d_in[13];
  const float* lin_b = (const float*)d_in[14];
  const int*   eidx  = (const int*)d_in[15];
  const int*   batch = (const int*)d_in[16];

  const int N = in_sizes[0] / HDIM;     // 10000
  const int E = in_sizes[1] / HDIM;     // 320000
  const int L = in_sizes[4];            // 4
  const int* src = eidx;
  const int* dst = eidx + E;

  const size_t NH = (size_t)N * HDIM;
  float* h    = (float*)d_ws;
  float* rbuf = h    + NH;
  float* hc   = rbuf + NH;
  float* mx   = hc   + NH;
  float* den  = mx   + NH;
  float* num  = den  + NH;
  float* t1   = num  + NH;              // N x 2H

  float* out    = (float*)d_out;
  float* pooled = out + NUM_GRAPHS * NUM_CLASSES;

  const dim3 gblk(128);
  const int eBlocks = (E + 7) / 8;                 // 1 wave per edge, 8 waves/block
  const int elBlocks = (int)((NH + 255) / 256);
  const int lnBlocks = (N + 7) / 8;                // 1 wave per row, 8 rows/block

  // encoder: h = x @ enc_W + enc_b
  gemm16_wmma_f32<<<dim3(N / 16, HDIM / 64), gblk, 0, stream>>>(x, enc_W, enc_b, h, HDIM, HDIM, 0);

  for (int layer = 0; layer < L; ++layer) {
    const float* rin = h;
    if (layer > 0) {
      ln_relu_kernel<<<lnBlocks, 256, 0, stream>>>(h, rbuf, ln_g + layer * HDIM, ln_b + layer * HDIM, N, HDIM);
      rin = rbuf;
    }
    init_seg_kernel<<<elBlocks, 256, 0, stream>>>(mx, den, num, (int)NH);
    edge_max_kernel<<<eBlocks, 256, 0, stream>>>(rin, ea, src, dst, t, layer, mx, E);
    edge_sum_kernel<<<eBlocks, 256, 0, stream>>>(rin, ea, src, dst, t, layer, mx, den, num, E);
    combine_kernel<<<elBlocks, 256, 0, stream>>>(rin, num, den, hc, (int)NH);
    // MLP: t1 = hc @ W1 + b1 ; LN+ReLU ; h (+)= t1 @ W2 + b2 (fused residual)
    gemm16_wmma_f32<<<dim3(N / 16, HDIM2 / 64), gblk, 0, stream>>>(
        hc, W1 + (size_t)layer * HDIM * HDIM2, b1 + layer * HDIM2, t1, HDIM, HDIM2, 0);
    ln_relu_kernel<<<lnBlocks, 256, 0, stream>>>(t1, t1, mlp_g + layer * HDIM2, mlp_b + layer * HDIM2, N, HDIM2);
    gemm16_wmma_f32<<<dim3(N / 16, HDIM / 64), gblk, 0, stream>>>(
        t1, W2 + (size_t)layer * HDIM2 * HDIM, b2 + layer * HDIM, h, HDIM2, HDIM, layer > 0 ? 1 : 0);
  }

  // final norm/act with layer-0 params, pool, classify
  ln_relu_kernel<<<lnBlocks, 256, 0, stream>>>(h, rbuf, ln_g, ln_b, N, HDIM);
  zero_kernel<<<(NUM_GRAPHS * HDIM + 255) / 256, 256, 0, stream>>>(pooled, NUM_GRAPHS * HDIM);
  pool_kernel<<<(int)((NH + 255) / 256), 256, 0, stream>>>(rbuf, batch, pooled, N);
  final_linear_kernel<<<(NUM_GRAPHS * NUM_CLASSES + 63) / 64, 64, 0, stream>>>(pooled, lin_W, lin_b, out);
}